// FFTBias2d_76081050681881
// MI455X (gfx1250) — compile-verified
//
#include <hip/hip_runtime.h>

#define B_  8
#define H_  16
#define E_  64
#define SHP 64            // shape_
#define S_  (SHP*SHP)     // 4096
#define NW  127

typedef __attribute__((ext_vector_type(2))) float v2f;
typedef __attribute__((ext_vector_type(4))) float f4;
typedef __attribute__((ext_vector_type(8))) float v8f;

// ---------------------------------------------------------------------------
// K1: streaming reduction of v -> v_m partials (per s1-group) and u_m.
// block = (b, h, g) with g = s1 quarter; 256 threads = (r 0..15) x (eq 0..15).
//   r  : s2 residue class (s2 mod 16) -> v_m partials exclusive in registers
//   eq : e quad (e = eq*4)            -> b128 fully-coalesced loads
// Reads v exactly once (134 MB) with non-temporal hint. Deterministic.
// ---------------------------------------------------------------------------
__global__ __launch_bounds__(256) void k1_reduce(const float* __restrict__ v,
                                                 float* __restrict__ vmp,   // [4][B][H][s2][e]
                                                 float* __restrict__ um) {  // [B][H][s1][e]
    const int bhg = blockIdx.x;
    const int g   = bhg & 3;
    const int bh  = bhg >> 2;
    const int h   = bh & (H_ - 1);
    const int b   = bh >> 4;
    const int tid = threadIdx.x;
    const int eq  = tid & 15;            // e quad
    const int r   = tid >> 4;            // s2 mod 16

    __shared__ f4 u_sh[16][16][16];      // [r][s1l][eq] u_m partials (64 KB)

    f4 vmacc[4];
#pragma unroll
    for (int k = 0; k < 4; ++k) vmacc[k] = (f4)0.f;

    const f4* v4 = (const f4*)v;
    const long base4 = ((long)b * S_) * (H_ * E_ / 4) + h * (E_ / 4) + eq;

    f4 uacc = (f4)0.f;
#pragma unroll 4
    for (int it = 0; it < 64; ++it) {
        const int s1l = it >> 2;                     // advances every 4 iters
        const int s2  = (it & 3) * 16 + r;
        const int s1  = g * 16 + s1l;
        const f4 val = __builtin_nontemporal_load(
            v4 + base4 + (long)(s1 * SHP + s2) * (H_ * E_ / 4));
        vmacc[it & 3] += val;                        // k = s2>>4, exclusive slot
        uacc += val;
        if ((it & 3) == 3) {                         // s1l complete for this thread
            u_sh[r][s1l][eq] = uacc;
            uacc = (f4)0.f;
        }
    }

    // write v_m partials: thread owns s2 in {r, r+16, r+32, r+48} x e-quad eq
    f4* vmp4 = (f4*)vmp;
    const long vmb4 = (((long)g * B_ + b) * H_ + h) * (SHP * E_ / 4);
#pragma unroll
    for (int k = 0; k < 4; ++k) {
        const int s2 = k * 16 + r;
        vmp4[vmb4 + s2 * (E_ / 4) + eq] = vmacc[k];
    }

    __syncthreads();

    // deterministic 16-way reduce of u partials; thread t -> (s1l, eq2)
    const int s1l = tid >> 4;
    const int eq2 = tid & 15;
    f4 s = (f4)0.f;
#pragma unroll
    for (int rr = 0; rr < 16; ++rr) s += u_sh[rr][s1l][eq2];
    f4* um4 = (f4*)um;
    um4[(((long)b * H_ + h) * SHP + (g * 16 + s1l)) * (E_ / 4) + eq2] = s;
}

// ---------------------------------------------------------------------------
// K2: per (b,h): build Toeplitz T_h in LDS, finish v_m reduction, then
// RxV = T*v_m and RxU = T*u_m as 64x64x64 fp32 matmuls via
// v_wmma_f32_16x16x4_f32. 8 waves x 4 tile-jobs = 32 output tiles.
// ---------------------------------------------------------------------------
__global__ __launch_bounds__(256) void k2_wmma(const float* __restrict__ w,
                                               const float* __restrict__ vmp,
                                               const float* __restrict__ um,
                                               float* __restrict__ rxv,   // [B][s2][H][e]
                                               float* __restrict__ rxu) { // [B][s1][H][e]
    const int bh  = blockIdx.x;
    const int h   = bh & (H_ - 1);
    const int b   = bh >> 4;
    const int tid = threadIdx.x;

    __shared__ float T_sh[SHP][SHP];   // T[t][j] = w[h][(t-j+64) mod 127]
    __shared__ float vm_sh[SHP][SHP];  // [j=s2][e]
    __shared__ float um_sh[SHP][SHP];  // [j=s1][e]

    // Toeplitz gather (scalar; 4 KB reused data, L2-hot)
#pragma unroll 4
    for (int i = 0; i < 16; ++i) {
        const int idx = tid + i * 256;           // 0..4095
        const int t = idx >> 6, j = idx & 63;
        T_sh[t][j] = w[h * NW + ((t - j + 64) % NW)];
    }

    // finish v_m reduction + load u_m, b128-wide
    const f4* vmp4 = (const f4*)vmp;
    const f4* um4  = (const f4*)um;
    f4* vm4 = (f4*)&vm_sh[0][0];
    f4* uu4 = (f4*)&um_sh[0][0];
    const long pbase4 = ((long)b * H_ + h) * (SHP * E_ / 4);
#pragma unroll
    for (int i = 0; i < 4; ++i) {
        const int idx = tid + i * 256;           // f4 index 0..1023
        f4 s = (f4)0.f;
#pragma unroll
        for (int g = 0; g < 4; ++g)
            s += vmp4[(long)g * (B_ * H_ * SHP * E_ / 4) + pbase4 + idx];
        vm4[idx] = s;
        uu4[idx] = um4[pbase4 + idx];
    }
    __syncthreads();

    const int wave = tid >> 5;
    const int lane = tid & 31;
    const int m    = lane & 15;          // row (A) / col (B,D) within tile
    const int kh   = lane >> 4;          // K half select per ISA 16x4 fp32 layout

    for (int job_i = 0; job_i < 4; ++job_i) {
        const int job  = wave * 4 + job_i;       // 0..31 (uniform per wave)
        const int mat  = job >> 4;               // 0: v_m -> RxV, 1: u_m -> RxU
        const int tile = job & 15;
        const int tBlk = tile >> 2;
        const int eBlk = tile & 3;
        const float (*src)[SHP] = mat ? um_sh : vm_sh;

        v8f acc = {0.f, 0.f, 0.f, 0.f, 0.f, 0.f, 0.f, 0.f};
#pragma unroll
        for (int ks = 0; ks < 16; ++ks) {        // K = 64 in steps of 4
            const int kb = ks * 4 + 2 * kh;      // lanes 0-15: K 0,1; 16-31: K 2,3
            v2f a, bb;
            a.x  = T_sh[tBlk * 16 + m][kb];
            a.y  = T_sh[tBlk * 16 + m][kb + 1];
            bb.x = src[kb][eBlk * 16 + m];
            bb.y = src[kb + 1][eBlk * 16 + m];
            acc = __builtin_amdgcn_wmma_f32_16x16x4_f32(
                      false, a, false, bb, (short)0, acc, false, false);
        }

        float* dst = mat ? rxu : rxv;
        const int col    = eBlk * 16 + m;
        const int rowoff = tBlk * 16 + (kh ? 8 : 0);   // D: lanes 16-31 hold M+8
#pragma unroll
        for (int rr = 0; rr < 8; ++rr) {
            const int row = rowoff + rr;
            dst[(((long)b * SHP + row) * H_ + h) * E_ + col] = acc[rr];
        }
    }
}

// ---------------------------------------------------------------------------
// K3: out[b, s1*64+s2, h, e] = RxV[b,s2,h,e] + RxU[b,s1,h,e].
// block = (b, s1); float4 outer-sum writeback; NT stores keep L2 for RxV reuse.
// ---------------------------------------------------------------------------
__global__ __launch_bounds__(256) void k3_out(const float* __restrict__ rxv,
                                              const float* __restrict__ rxu,
                                              float* __restrict__ out) {
    const int bs1 = blockIdx.x;
    const int s1  = bs1 & 63;
    const int b   = bs1 >> 6;
    const int tid = threadIdx.x;

    __shared__ f4 rxu_sh[256];       // RxU row for this (b,s1): [h][e] quads
    rxu_sh[tid] = ((const f4*)(rxu + (((long)b * SHP + s1) * H_) * E_))[tid];
    __syncthreads();

    const f4* rxv4 = (const f4*)(rxv + (long)b * SHP * H_ * E_);
    f4*       out4 = (f4*)out + ((long)b * S_ + s1 * SHP) * (H_ * E_ / 4);

#pragma unroll 4
    for (int it = 0; it < 64; ++it) {
        const int flat = tid + it * 256;   // 0..16383 f4 elements
        const int s2   = flat >> 8;
        const int rem  = flat & 255;       // (h,e) quad, contiguous across lanes
        const f4 o = rxv4[s2 * 256 + rem] + rxu_sh[rem];
        __builtin_nontemporal_store(o, out4 + s2 * 256 + rem);
    }
}

// ---------------------------------------------------------------------------
extern "C" void kernel_launch(void* const* d_in, const int* in_sizes, int n_in,
                              void* d_out, int out_size, void* d_ws, size_t ws_size,
                              hipStream_t stream) {
    const float* v = (const float*)d_in[0];   // (8, 4096, 16, 64) fp32
    const float* w = (const float*)d_in[1];   // (1, 16, 127)      fp32
    float* ws = (float*)d_ws;

    // workspace layout (floats): 14 MB total
    float* vmp = ws;                 // 4*B*H*64*64 = 2,097,152
    float* um  = ws + 2097152;       //   B*H*64*64 =   524,288
    float* rxv = ws + 2621440;       //   B*64*H*64 =   524,288
    float* rxu = ws + 3145728;       //   B*64*H*64 =   524,288
    float* out = (float*)d_out;

    k1_reduce<<<B_ * H_ * 4, 256, 0, stream>>>(v, vmp, um);
    k2_wmma  <<<B_ * H_,     256, 0, stream>>>(w, vmp, um, rxv, rxu);
    k3_out   <<<B_ * SHP,    256, 0, stream>>>(rxv, rxu, out);
}